// slice_84198538871183
// MI455X (gfx1250) — compile-verified
//
#include <hip/hip_runtime.h>
#include <hip/hip_bf16.h>

// CDNA5 / gfx1250, wave32. WMMA fp32 (16x16x4) used as a full-precision
// cross-lane reduction engine: with B = ones, D[m][n] = sum_k A[m][k], so one
// WMMA collapses 32 lane partials to two half-wave sums; a single shfl_xor(16)
// finishes the wave32 reduction.

typedef __attribute__((ext_vector_type(2))) float v2f;
typedef __attribute__((ext_vector_type(8))) float v8f;

__device__ __forceinline__ float wave_sum_wmma(float x) {
  v2f a; a[0] = x;    a[1] = 0.0f;
  v2f b; b[0] = 1.0f; b[1] = 1.0f;
  v8f c = {};
  // D = A(16x4 f32) * B(4x16 ones) + 0  -> row sums of A replicated across N
  c = __builtin_amdgcn_wmma_f32_16x16x4_f32(false, a, false, b,
                                            (short)0, c, false, false);
  float s = c[0] + c[1] + c[2] + c[3] + c[4] + c[5] + c[6] + c[7];
  // lanes 0-15 hold sum over rows 0-7 (x0..7 + x16..23),
  // lanes 16-31 hold sum over rows 8-15 (x8..15 + x24..31)
  s += __shfl_xor(s, 16, 32);
  return s;  // full wave sum in every lane
}

__global__ void __launch_bounds__(256)
zero_kernel(float4* __restrict__ p, int n4) {
  int i = blockIdx.x * blockDim.x + threadIdx.x;
  int stride = gridDim.x * blockDim.x;
  float4 z = make_float4(0.f, 0.f, 0.f, 0.f);
  for (; i < n4; i += stride) p[i] = z;
}

// Per-(batch, component) sums of pred; one thread per vertex row (float4).
// B == 2, N % 32 == 0  ->  batch index is a single wave-uniform compare.
__global__ void __launch_bounds__(256)
mean_kernel(const float4* __restrict__ pred,
            float* __restrict__ meanSum, int N, int BN) {
  int g = blockIdx.x * blockDim.x + threadIdx.x;
  float px = 0.f, py = 0.f, pz = 0.f, pw = 0.f;
  bool valid = (g < BN);
  if (valid) {
    float4 p = pred[g];
    px = p.x; py = p.y; pz = p.z; pw = p.w;
  }
  float sx = wave_sum_wmma(px);
  float sy = wave_sum_wmma(py);
  float sz = wave_sum_wmma(pz);
  float sw = wave_sum_wmma(pw);
  if (((threadIdx.x & 31) == 0) && valid) {
    int b = (g >= N) ? 1 : 0;  // wave-uniform
    atomicAdd(&meanSum[b * 4 + 0], sx);
    atomicAdd(&meanSum[b * 4 + 1], sy);
    atomicAdd(&meanSum[b * 4 + 2], sz);
    atomicAdd(&meanSum[b * 4 + 3], sw);
  }
}

// One thread per tetra: gather 4 centered verts (L2-resident), scatter the
// Laplacian contributions + counts (L2 atomics), compute 6 edges -> pts/mask
// streams + L2/L3 partials (WMMA wave reduce, one atomic per wave).
__global__ void __launch_bounds__(256)
tetra_kernel(const float4* __restrict__ pred,
             const int4* __restrict__ tetra,
             const float* __restrict__ meanSum,
             float* __restrict__ term,
             float* __restrict__ cnt,
             float* __restrict__ lsum,
             float* __restrict__ outPts,
             float* __restrict__ outMask,
             float invN, int N, int T, int BT) {
  int g = blockIdx.x * blockDim.x + threadIdx.x;
  float l2p = 0.0f, l3p = 0.0f;
  if (g < BT) {
    bool b1 = (g >= T);  // batch index (B == 2), wave-uniform
    // Both batches' mean sums come from uniform addresses -> scalar loads.
    float4 mA = *(const float4*)(meanSum);
    float4 mB = *(const float4*)(meanSum + 4);
    float mnx = (b1 ? mB.x : mA.x) * invN;
    float mny = (b1 ? mB.y : mA.y) * invN;
    float mnz = (b1 ? mB.z : mA.z) * invN;
    float mnw = (b1 ? mB.w : mA.w) * invN;

    const float4* pb4 = pred + (b1 ? (size_t)N : (size_t)0);
    int4 id = tetra[g];
    int idx[4] = {id.x, id.y, id.z, id.w};
    float vx[4], vy[4], vz[4], vw[4];
#pragma unroll
    for (int j = 0; j < 4; ++j) {
      float4 p = pb4[idx[j]];
      vx[j] = p.x - mnx; vy[j] = p.y - mny;
      vz[j] = p.z - mnz; vw[j] = p.w - mnw;
    }
    float sx = vx[0] + vx[1] + vx[2] + vx[3];
    float sy = vy[0] + vy[1] + vy[2] + vy[3];
    float sz = vz[0] + vz[1] + vz[2] + vz[3];
    float sw = vw[0] + vw[1] + vw[2] + vw[3];

    size_t tb = b1 ? (size_t)N : (size_t)0;
#pragma unroll
    for (int j = 0; j < 4; ++j) {
      float* tp = term + (tb + (size_t)idx[j]) * 4;
      atomicAdd(tp + 0, sx - 4.0f * vx[j]);
      atomicAdd(tp + 1, sy - 4.0f * vy[j]);
      atomicAdd(tp + 2, sz - 4.0f * vz[j]);
      atomicAdd(tp + 3, sw - 4.0f * vw[j]);
      atomicAdd(cnt + tb + (size_t)idx[j], 1.0f);
    }

    const int EA[6] = {0, 0, 0, 1, 1, 2};
    const int EB[6] = {1, 2, 3, 2, 3, 3};
    float* pp = outPts + (size_t)g * 18;   // (b*T+t, 6, 3), fully coalesced
    float* mm = outMask + (size_t)g * 6;
#pragma unroll
    for (int e = 0; e < 6; ++e) {
      int a = EA[e], c = EB[e];
      float dx = vx[a] - vx[c], dy = vy[a] - vy[c];
      float dz = vz[a] - vz[c], dw = vw[a] - vw[c];
      float wa = vw[a], wb = vw[c];
      float ed = wa * wb;                      // ALPHA == 0
      l2p += ed;
      float nr = sqrtf(dx * dx + dy * dy + dz * dz + dw * dw);
      float df = nr - 0.4f;
      l3p += df * df;
      float q = dw;
      float sq = (fabsf(q) > 1e-12f) ? q : 1.0f;
      float t = (0.0f - wa) / sq;
      bool m = ed < 0.0f;
      pp[e * 3 + 0] = m ? (vx[a] + t * dx) : 0.0f;
      pp[e * 3 + 1] = m ? (vy[a] + t * dy) : 0.0f;
      pp[e * 3 + 2] = m ? (vz[a] + t * dz) : 0.0f;
      mm[e] = m ? 1.0f : 0.0f;
    }
  }
  // reconverged: all 256 lanes alive, EXEC all-ones for WMMA
  float s2 = wave_sum_wmma(l2p);
  float s3 = wave_sum_wmma(l3p);
  if ((threadIdx.x & 31) == 0) {
    atomicAdd(&lsum[1], s2);
    atomicAdd(&lsum[2], s3);
  }
}

// Per-vertex: term /= max(3*count,1); accumulate sum(term^2) -> L1.
__global__ void __launch_bounds__(256)
term_kernel(const float4* __restrict__ term,
            const float* __restrict__ cnt,
            float* __restrict__ lsum, int BN) {
  int g = blockIdx.x * blockDim.x + threadIdx.x;
  float p = 0.0f;
  if (g < BN) {
    float4 t = term[g];
    float c = cnt[g];
    float inv = 1.0f / fmaxf(3.0f * c, 1.0f);
    float tx = t.x * inv, ty = t.y * inv, tz = t.z * inv, tw = t.w * inv;
    p = tx * tx + ty * ty + tz * tz + tw * tw;
  }
  float s = wave_sum_wmma(p);
  if ((threadIdx.x & 31) == 0) atomicAdd(&lsum[0], s);
}

__global__ void finalize_kernel(const float* __restrict__ lsum,
                                float* __restrict__ out, int BN, int BT) {
  if (threadIdx.x == 0) {
    out[0] = lsum[0] / ((float)BN * 4.0f);  // mean over batches of L1
    out[1] = lsum[1] / ((float)BT * 6.0f);  // L2
    out[2] = lsum[2] / ((float)BT * 6.0f);  // L3
    out[3] = 0.0f;
  }
}

extern "C" void kernel_launch(void* const* d_in, const int* in_sizes, int n_in,
                              void* d_out, int out_size, void* d_ws, size_t ws_size,
                              hipStream_t stream) {
  const float* pred = (const float*)d_in[0];  // (B, N, 4) fp32
  const int* tetra = (const int*)d_in[1];     // (B, T, 4) int32
  const int B  = 2;
  const int BN = in_sizes[0] / 4;  // B*N = 240000 (multiple of 32)
  const int BT = in_sizes[1] / 4;  // B*T = 1200000 (multiple of 32)
  const int N  = BN / B;
  const int T  = BT / B;

  // Workspace layout (floats): [0..7] meanSum, [8..10] L1/L2/L3 sums,
  // [16 ..) term (B*N*4), then cnt (B*N).  ~4.8 MB total.
  float* wsf     = (float*)d_ws;
  float* meanSum = wsf;
  float* lsum    = wsf + 8;
  float* term    = wsf + 16;
  float* cnt     = term + (size_t)BN * 4;
  int nzero  = 16 + BN * 4 + BN;      // 1,200,016 floats
  int nzero4 = (nzero + 3) / 4;       // float4 stores

  float* out     = (float*)d_out;
  float* outPts  = out + 4;
  float* outMask = out + 4 + (size_t)BT * 18;

  zero_kernel<<<512, 256, 0, stream>>>((float4*)wsf, nzero4);
  mean_kernel<<<(BN + 255) / 256, 256, 0, stream>>>(
      (const float4*)pred, meanSum, N, BN);
  tetra_kernel<<<(BT + 255) / 256, 256, 0, stream>>>(
      (const float4*)pred, (const int4*)tetra, meanSum, term, cnt, lsum,
      outPts, outMask, (float)(1.0 / (double)N), N, T, BT);
  term_kernel<<<(BN + 255) / 256, 256, 0, stream>>>(
      (const float4*)term, cnt, lsum, BN);
  finalize_kernel<<<1, 32, 0, stream>>>(lsum, out, BN, BT);
}